// BetaTCVAE_26242250179260
// MI455X (gfx1250) — compile-verified
//
#include <hip/hip_runtime.h>

#define BETA 6.0f
#define LOG_2PI 1.8378770664093453f
#define B_N 2048
#define D_N 64
#define NTILES (B_N / 16)   // 128 tiles of 16 rows

typedef float v2f __attribute__((ext_vector_type(2)));
typedef float v8f __attribute__((ext_vector_type(8)));

// -------------------------------------------------------------------------
// Kernel 1: per-(j,d) coefficients for the rank-3 expansion of the Gaussian
// log-density:  P_d[i,j] = b1[j,d]*z_i^2 + b2[j,d]*z_i + c[j,d]
// Written in BOTH layouts: row-major [j][d] (kernel 2 streams rows) and
// transposed [d][j] (kernel 3 needs lane-contiguous j at fixed d).
// -------------------------------------------------------------------------
__global__ void __launch_bounds__(256)
btcvae_prep(const float* __restrict__ z_mean,
            const float* __restrict__ z_logvar,
            const float* __restrict__ z_samp,
            float* __restrict__ z2,
            float* __restrict__ b1,
            float* __restrict__ b2,
            float* __restrict__ cjd,
            float* __restrict__ b1T,
            float* __restrict__ b2T,
            float* __restrict__ cT) {
  int e = blockIdx.x * blockDim.x + threadIdx.x;
  if (e >= B_N * D_N) return;
  int j = e / D_N;
  int d = e - j * D_N;
  float lv = z_logvar[e];
  float m  = z_mean[e];
  float z  = z_samp[e];
  float w  = __expf(-lv);            // exp(-log_var)
  float v1 = -0.5f * w;
  float v2 = m * w;
  float vc = -0.5f * (m * m * w + lv + LOG_2PI);
  z2[e]  = z * z;
  b1[e]  = v1;
  b2[e]  = v2;
  cjd[e] = vc;
  int et = d * B_N + j;              // transposed [d][j]
  b1T[et] = v1;
  b2T[et] = v2;
  cT[et]  = vc;
}

// -------------------------------------------------------------------------
// Kernel 2: S[i,j] = sum_d P_d[i,j] via f32 WMMA GEMM (K = 64 d's x 4),
// then logsumexp over j  ->  log_qz[i].
// Tile orientation: C[M=j_local, N=i_local], so each lane's 8 accumulator
// VGPRs hold 8 j-values of one column i -> the j-reduction is lane-local.
// One workgroup per i-tile, 8 waves split the 128 j-tiles.
// -------------------------------------------------------------------------
__global__ void __launch_bounds__(256)
btcvae_logqz(const float* __restrict__ z,
             const float* __restrict__ z2,
             const float* __restrict__ b1,
             const float* __restrict__ b2,
             const float* __restrict__ cjd,
             float* __restrict__ log_qz) {
  const int itile = blockIdx.x;
  const int tid   = threadIdx.x;
  const int lane  = tid & 31;
  const int wave  = tid >> 5;
  const bool hi   = (lane >= 16);
  const int l16   = lane & 15;
  const int i     = itile * 16 + l16;

  // B-operand (indexed by i only) is identical for all 8 waves: cache in LDS.
  __shared__ float bcache[D_N][32][2];
  __shared__ float red[8][32][2];

  if (wave == 0) {
    for (int d = 0; d < D_N; ++d) {
      float tz2 = z2[i * D_N + d];
      float tz  = z[i * D_N + d];
      bcache[d][lane][0] = hi ? 1.0f : tz2;   // K rows (z^2, z, 1, 0)
      bcache[d][lane][1] = hi ? 0.0f : tz;
    }
  }
  __syncthreads();

  float mS = -__builtin_inff();   // running max  (logsumexp over j)
  float sS = 0.0f;                // running sum of exp

  const float* abase0 = hi ? cjd : b1;   // lane-half select of A K-components

  for (int jt = wave; jt < NTILES; jt += 8) {
    const int j = jt * 16 + l16;
    v8f acc = {};
    const float* arow = abase0 + (size_t)j * D_N;
    const float* brow = b2 + (size_t)j * D_N;
    __builtin_prefetch(arow + 8 * 16 * D_N, 0, 1);
    for (int d = 0; d < D_N; ++d) {
      v2f a, b;
      a.x = arow[d];                       // b1[j,d] (lo lanes) / c[j,d] (hi)
      float t = brow[d];
      a.y = hi ? 0.0f : t;                 // b2[j,d] / 0
      b.x = bcache[d][lane][0];
      b.y = bcache[d][lane][1];
      acc = __builtin_amdgcn_wmma_f32_16x16x4_f32(false, a, false, b,
                                                  (short)0, acc, false, false);
    }
    // fold this tile's 8 j-values into the running logsumexp (lane-local)
    float m8 = acc[0];
#pragma unroll
    for (int r = 1; r < 8; ++r) m8 = fmaxf(m8, acc[r]);
    float nm = fmaxf(mS, m8);
    float s  = sS * __expf(mS - nm);
#pragma unroll
    for (int r = 0; r < 8; ++r) s += __expf(acc[r] - nm);
    mS = nm; sS = s;
  }

  red[wave][lane][0] = mS;
  red[wave][lane][1] = sS;
  __syncthreads();

  if (tid < 16) {                // combine 8 waves x 2 lane-halves per i
    float m = -__builtin_inff(), s = 0.0f;
    for (int w = 0; w < 8; ++w)
      for (int h = 0; h < 2; ++h) {
        float mk = red[w][tid + 16 * h][0];
        float sk = red[w][tid + 16 * h][1];
        float nm = fmaxf(m, mk);
        s = s * __expf(m - nm) + sk * __expf(mk - nm);
        m = nm;
      }
    log_qz[itile * 16 + tid] = m + __logf(s);
  }
}

// -------------------------------------------------------------------------
// Kernel 3: per-dimension logsumexp over j: lse_jd[i,d] = LSE_j P_d[i,j].
// Grid (128 i-tiles x 64 d). One WMMA (C=0) per j-tile; online LSE in regs.
// Uses the TRANSPOSED [d][j] coefficient arrays so lane loads at fixed d
// are contiguous in j (coalesced), since this kernel touches all 268M
// pairwise elements and dominates runtime.
// -------------------------------------------------------------------------
__global__ void __launch_bounds__(128)
btcvae_lsejd(const float* __restrict__ z,
             const float* __restrict__ z2,
             const float* __restrict__ b1T,
             const float* __restrict__ b2T,
             const float* __restrict__ cT,
             float* __restrict__ lse_jd) {
  const int itile = blockIdx.x;
  const int d     = blockIdx.y;
  const int tid   = threadIdx.x;
  const int lane  = tid & 31;
  const int wave  = tid >> 5;
  const bool hi   = (lane >= 16);
  const int l16   = lane & 15;
  const int i     = itile * 16 + l16;

  v2f b;
  {
    float tz2 = z2[i * D_N + d];
    float tz  = z[i * D_N + d];
    b.x = hi ? 1.0f : tz2;
    b.y = hi ? 0.0f : tz;
  }

  float mS = -__builtin_inff();
  float sS = 0.0f;
  const float* arow0 = (hi ? cT : b1T) + (size_t)d * B_N;  // [d][j] row
  const float* arow1 = b2T + (size_t)d * B_N;

  for (int jt = wave; jt < NTILES; jt += 4) {
    const int j = jt * 16 + l16;
    __builtin_prefetch(arow0 + j + 4 * 16, 0, 1);
    v2f a;
    a.x = arow0[j];                  // contiguous across the half-wave
    float t = arow1[j];
    a.y = hi ? 0.0f : t;
    v8f c = {};
    c = __builtin_amdgcn_wmma_f32_16x16x4_f32(false, a, false, b,
                                              (short)0, c, false, false);
    float m8 = c[0];
#pragma unroll
    for (int r = 1; r < 8; ++r) m8 = fmaxf(m8, c[r]);
    float nm = fmaxf(mS, m8);
    float s  = sS * __expf(mS - nm);
#pragma unroll
    for (int r = 0; r < 8; ++r) s += __expf(c[r] - nm);
    mS = nm; sS = s;
  }

  __shared__ float red[4][32][2];
  red[wave][lane][0] = mS;
  red[wave][lane][1] = sS;
  __syncthreads();

  if (tid < 16) {                // 4 waves x 2 lane-halves per i
    float m = -__builtin_inff(), s = 0.0f;
    for (int w = 0; w < 4; ++w)
      for (int h = 0; h < 2; ++h) {
        float mk = red[w][tid + 16 * h][0];
        float sk = red[w][tid + 16 * h][1];
        float nm = fmaxf(m, mk);
        s = s * __expf(m - nm) + sk * __expf(mk - nm);
        m = nm;
      }
    lse_jd[(size_t)(itile * 16 + tid) * D_N + d] = m + __logf(s);
  }
}

// -------------------------------------------------------------------------
// Kernel 4: deterministic final reduction:
//   out = (BETA-1) * mean_i(log_qz[i] - sum_d lse_jd[i,d]) + sum(kl)
// -------------------------------------------------------------------------
__global__ void __launch_bounds__(256)
btcvae_final(const float* __restrict__ kl,
             const float* __restrict__ log_qz,
             const float* __restrict__ lse_jd,
             float* __restrict__ out) {
  const int tid = threadIdx.x;
  float kl_sum = 0.0f;
  for (int e = tid; e < B_N * D_N; e += 256) kl_sum += kl[e];
  float t2 = 0.0f;                        // sum over all (i,d) of lse_jd
  for (int e = tid; e < B_N * D_N; e += 256) t2 += lse_jd[e];
  float t1 = 0.0f;                        // sum over i of log_qz
  for (int e = tid; e < B_N; e += 256) t1 += log_qz[e];

  float partial = kl_sum + ((BETA - 1.0f) / (float)B_N) * (t1 - t2);

  __shared__ float red[256];
  red[tid] = partial;
  __syncthreads();
  for (int s = 128; s > 0; s >>= 1) {
    if (tid < s) red[tid] += red[tid + s];
    __syncthreads();
  }
  if (tid == 0) out[0] = red[0];
}

// -------------------------------------------------------------------------
extern "C" void kernel_launch(void* const* d_in, const int* in_sizes, int n_in,
                              void* d_out, int out_size, void* d_ws, size_t ws_size,
                              hipStream_t stream) {
  const float* kl       = (const float*)d_in[0];
  const float* z_mean   = (const float*)d_in[1];
  const float* z_logvar = (const float*)d_in[2];
  const float* z_samp   = (const float*)d_in[3];
  float* out = (float*)d_out;

  const size_t ND = (size_t)B_N * D_N;     // 131072
  float* ws     = (float*)d_ws;
  float* z2     = ws;                       // [B_N*D_N]  row-major
  float* b1     = z2  + ND;                 // [B_N*D_N]  row-major
  float* b2     = b1  + ND;                 // [B_N*D_N]  row-major
  float* cjd    = b2  + ND;                 // [B_N*D_N]  row-major
  float* b1T    = cjd + ND;                 // [D_N*B_N]  transposed
  float* b2T    = b1T + ND;                 // [D_N*B_N]  transposed
  float* cT     = b2T + ND;                 // [D_N*B_N]  transposed
  float* log_qz = cT  + ND;                 // [B_N]
  float* lse_jd = log_qz + B_N;             // [B_N*D_N]
  (void)in_sizes; (void)n_in; (void)out_size; (void)ws_size;

  btcvae_prep<<<(int)((ND + 255) / 256), 256, 0, stream>>>(
      z_mean, z_logvar, z_samp, z2, b1, b2, cjd, b1T, b2T, cT);

  btcvae_logqz<<<NTILES, 256, 0, stream>>>(z_samp, z2, b1, b2, cjd, log_qz);

  btcvae_lsejd<<<dim3(NTILES, D_N), 128, 0, stream>>>(
      z_samp, z2, b1T, b2T, cT, lse_jd);

  btcvae_final<<<1, 256, 0, stream>>>(kl, log_qz, lse_jd, out);
}